// Interpolator_66400194396539
// MI455X (gfx1250) — compile-verified
//
#include <hip/hip_runtime.h>
#include <hip/hip_bf16.h>
#include <math.h>

// ---------------------------------------------------------------------------
// Problem constants (from the reference)
// ---------------------------------------------------------------------------
#define N_SAMPLES 4096
#define N_CORE    4094   // N_SAMPLES - 2
#define N_NEW     1000000

typedef float v2f __attribute__((ext_vector_type(2)));
typedef float v8f __attribute__((ext_vector_type(8)));

static __device__ __forceinline__ v8f wmma_f32_k4(v2f a, v2f b, v8f c) {
    // 8 args: (neg_a, A, neg_b, B, c_mod, C, reuse_a, reuse_b)
    return __builtin_amdgcn_wmma_f32_16x16x4_f32(false, a, false, b,
                                                 (short)0, c, false, false);
}

// ---------------------------------------------------------------------------
// Fused matvec + bias + activation via V_WMMA_F32_16X16X4_F32.
//
//   out[m] = act( sum_k W[m,k]*x[k] + b[m] ),  W row-major, ld == K.
//
// 4 waves / block, one 16-row block per workgroup.  K-groups of 4 are
// interleaved across waves; the dual-chain main loop covers groups
// wave + {0,4} + 8*i for i in [0, ndual) with ndual = (K/4) >> 3 --
// UNIFORM across waves, so the loop bound is scalar (no exec juggling).
//
// Branch-free inner loop:
//  * Only D[:,0] is consumed and it depends only on B column 0 / A row m,
//    so B is x broadcast to ALL lanes (every D column computes the same
//    dot product) and OOB A rows are clamped to M-1 (never stored).
//  * A (ISA 32-bit 16x4 layout): lanes 0-15 -> M=lane, VGPR0/1=K0/K1;
//    lanes 16-31 -> M=lane-16, VGPR0/1=K2/K3  => one b64 load per lane.
//  * W streamed exactly once -> non-temporal loads (don't thrash 192MB L2).
//  * Two accumulator chains (c0/c1) so loads + WMMAs pipeline under a
//    partial s_wait_loadcnt.
//  * Bumped pointers + immediate offsets instead of per-iteration 64-bit
//    address recomputation.
//
// D column 0 lands in lane 0 (M=0..7, VGPR 0..7) and lane 16 (M=8..15).
// Wave partials reduced through LDS; bias + activation fused in epilogue.
// act: 0 = relu, 1 = sigmoid()*(hi-lo)+lo
// ---------------------------------------------------------------------------
__global__ __launch_bounds__(128)
void mlp_matvec_wmma(const float* __restrict__ W,
                     const float* __restrict__ x,
                     const float* __restrict__ bias,
                     float* __restrict__ out,
                     int M, int K, int act, float lo, float hi) {
    __shared__ float partial[4][16];

    const int wave      = threadIdx.x >> 5;
    const int lane      = threadIdx.x & 31;
    const int row_block = blockIdx.x * 16;
    const int m_local   = lane & 15;
    const int k_lane    = (lane >> 4) << 1;          // 0 (lanes 0-15) or 2

    const int  row   = row_block + m_local;
    const int  row_c = (row < M) ? row : (M - 1);    // clamp, never masked
    const float* __restrict__ pW = W + (size_t)row_c * (size_t)K + k_lane;
    const float* __restrict__ pX = x + k_lane;

    v8f c0 = {0.f, 0.f, 0.f, 0.f, 0.f, 0.f, 0.f, 0.f};
    v8f c1 = {0.f, 0.f, 0.f, 0.f, 0.f, 0.f, 0.f, 0.f};

    const int ngroups = K >> 2;        // full groups of 4 k-values
    const int ndual   = ngroups >> 3;  // uniform dual-chain trip count

    // Bumped pointers: this wave starts at group `wave`; each iteration
    // consumes groups g and g+4 (the +4 group sits at +16 floats = +64 B).
    const char* pa = (const char*)(pW + (wave << 2));
    const char* pb = (const char*)(pX + (wave << 2));

    for (int i = 0; i < ndual; ++i) {
        v2f a0 = __builtin_nontemporal_load((const v2f*)pa);
        v2f b0 = *(const v2f*)pb;
        v2f a1 = __builtin_nontemporal_load((const v2f*)(pa + 64));
        v2f b1 = *(const v2f*)(pb + 64);
        pa += 128;                       // 8 groups * 4 floats
        pb += 128;
        c0 = wmma_f32_k4(a0, b0, c0);
        c1 = wmma_f32_k4(a1, b1, c1);
    }

    // Leftover full groups (ngroups % 8 of them, split across waves: at most
    // 2 iterations for any wave).
    for (int g = (ndual << 3) + wave; g < ngroups; g += 4) {
        v2f a = __builtin_nontemporal_load((const v2f*)(pW + (g << 2)));
        v2f b = *(const v2f*)(pX + (g << 2));
        c0 = wmma_f32_k4(a, b, c0);
    }

    // Partial tail group (K % 4 != 0; e.g. K==4094 -> 2 leftover columns),
    // handled by its owner wave.  Wave-uniform branch: EXEC is all-1s at the
    // WMMA; only the loads inside are lane-predicated.
    if ((K & 3) != 0 && wave == (ngroups & 3)) {
        const int k0 = ngroups << 2;
        const int kk = k0 + k_lane;
        float a0 = 0.f, a1 = 0.f, b0 = 0.f, b1 = 0.f;
        if (kk < K)     { a0 = pW[kk - k_lane];     b0 = x[kk];     }
        if (kk + 1 < K) { a1 = pW[kk + 1 - k_lane]; b1 = x[kk + 1]; }
        v2f a = {a0, a1};
        v2f b = {b0, b1};
        c0 = wmma_f32_k4(a, b, c0);
    }

    v8f c = c0 + c1;

    // Column 0 of D: lane 0 holds M=0..7 (VGPR 0..7), lane 16 holds M=8..15.
    if (lane == 0) {
        #pragma unroll
        for (int i = 0; i < 8; ++i) partial[wave][i] = c[i];
    }
    if (lane == 16) {
        #pragma unroll
        for (int i = 0; i < 8; ++i) partial[wave][8 + i] = c[i];
    }
    __syncthreads();

    if (threadIdx.x < 16) {
        const int r = row_block + threadIdx.x;
        if (r < M) {
            float s = partial[0][threadIdx.x] + partial[1][threadIdx.x] +
                      partial[2][threadIdx.x] + partial[3][threadIdx.x] +
                      bias[r];
            float v;
            if (act == 0) {
                v = fmaxf(s, 0.f);
            } else {
                float sg = 1.f / (1.f + __expf(-s));
                v = sg * (hi - lo) + lo;
            }
            out[r] = v;
        }
    }
}

// ---------------------------------------------------------------------------
// Build x_hat = cumsum(offsets) with the reference's fixups.
//   offsets[0]=0; offsets[i]=off_core[i-1] (i=1..4094); offsets[4095]=0;
//   offsets[4094] /= 2;  x_hat = cumsum(offsets);  x_hat[4095] = N_NEW-1.
// Single block of 256 threads, 16 elements each, LDS Hillis-Steele scan.
// ---------------------------------------------------------------------------
__global__ __launch_bounds__(256)
void build_xhat(const float* __restrict__ off_core,
                float* __restrict__ x_hat,
                float last_val) {
    __shared__ float sums[256];
    const int t = threadIdx.x;
    const int base = t * 16;

    float loc[16];
    float s = 0.f;
    #pragma unroll
    for (int j = 0; j < 16; ++j) {
        const int i = base + j;
        float o = 0.f;
        if (i > 0 && i < N_SAMPLES - 1) {
            o = off_core[i - 1];
            if (i == N_SAMPLES - 2) o *= 0.5f;
        }
        loc[j] = o;
        s += o;
    }
    sums[t] = s;
    __syncthreads();

    for (int d = 1; d < 256; d <<= 1) {
        float v   = sums[t];
        float add = (t >= d) ? sums[t - d] : 0.f;
        __syncthreads();
        sums[t] = v + add;
        __syncthreads();
    }
    float run = (t > 0) ? sums[t - 1] : 0.f;   // exclusive prefix

    #pragma unroll
    for (int j = 0; j < 16; ++j) {
        run += loc[j];
        const int i = base + j;
        x_hat[i] = (i == N_SAMPLES - 1) ? last_val : run;
    }
}

// ---------------------------------------------------------------------------
// 1M-point linear interpolation.  x_hat / y_hat cached in 32 KB of LDS;
// 12-step binary search per query (searchsorted left, ind=clip(pos-1,0,n-2)),
// 4 queries per thread.
// ---------------------------------------------------------------------------
__global__ __launch_bounds__(256)
void interp1d_kernel(const float* __restrict__ x_new,
                     const float* __restrict__ x_hat,
                     const float* __restrict__ y_hat,
                     float* __restrict__ y_pred,
                     int n_new) {
    __shared__ float sx[N_SAMPLES];
    __shared__ float sy[N_SAMPLES];

    for (int i = threadIdx.x; i < N_SAMPLES; i += 256) {
        sx[i] = x_hat[i];
        sy[i] = y_hat[i];
    }
    __syncthreads();

    const int base = blockIdx.x * 1024;
    #pragma unroll
    for (int j = 0; j < 4; ++j) {
        const int idx = base + j * 256 + threadIdx.x;
        if (idx < n_new) {
            const float xq = x_new[idx];
            int lo = 0, hi = N_SAMPLES;
            while (lo < hi) {
                const int mid = (lo + hi) >> 1;
                if (sx[mid] < xq) lo = mid + 1; else hi = mid;
            }
            int ind = lo - 1;
            ind = (ind < 0) ? 0 : ind;
            ind = (ind > N_SAMPLES - 2) ? (N_SAMPLES - 2) : ind;
            const float x0 = sx[ind], x1 = sx[ind + 1];
            const float y0 = sy[ind], y1 = sy[ind + 1];
            y_pred[idx] = y0 + (y1 - y0) / (x1 - x0) * (xq - x0);
        }
    }
}

// ---------------------------------------------------------------------------
// Host-side orchestration.  Stream order provides dependencies:
// layer1 -> layer2 -> scan -> interp.
// Inputs (setup_inputs order):
//  0 x_new  1 x_sampled(unused)  2 input_noise
//  3 w1_off 4 b1_off 5 w2_off 6 b2_off
//  7 w1_val 8 b1_val 9 w2_val 10 b2_val
// Output: [y_pred(1e6) | x_hat(4096) | y_hat(4096)]
// ---------------------------------------------------------------------------
extern "C" void kernel_launch(void* const* d_in, const int* in_sizes, int n_in,
                              void* d_out, int out_size, void* d_ws, size_t ws_size,
                              hipStream_t stream) {
    const float* x_new       = (const float*)d_in[0];
    const float* input_noise = (const float*)d_in[2];
    const float* w1_off = (const float*)d_in[3];
    const float* b1_off = (const float*)d_in[4];
    const float* w2_off = (const float*)d_in[5];
    const float* b2_off = (const float*)d_in[6];
    const float* w1_val = (const float*)d_in[7];
    const float* b1_val = (const float*)d_in[8];
    const float* w2_val = (const float*)d_in[9];
    const float* b2_val = (const float*)d_in[10];

    float* y_pred = (float*)d_out;
    float* x_hat  = y_pred + N_NEW;
    float* y_hat  = x_hat + N_SAMPLES;

    float* ws       = (float*)d_ws;
    float* h_val    = ws;                  // 4096
    float* h_off    = ws + N_SAMPLES;      // 4094 (padded slot of 4096)
    float* off_core = ws + 2 * N_SAMPLES;  // 4094

    const dim3 blk(128);
    const int grid_val = (N_SAMPLES + 15) / 16;  // 256
    const int grid_off = (N_CORE + 15) / 16;     // 256

    // Layer 1 (independent): h = relu(W1 @ x + b1)
    mlp_matvec_wmma<<<grid_val, blk, 0, stream>>>(w1_val, input_noise, b1_val,
                                                  h_val, N_SAMPLES, N_SAMPLES,
                                                  0, 0.f, 0.f);
    mlp_matvec_wmma<<<grid_off, blk, 0, stream>>>(w1_off, input_noise, b1_off,
                                                  h_off, N_CORE, N_CORE,
                                                  0, 0.f, 0.f);
    // Layer 2: sigmoid epilogues -> y_hat in [-1,1], off_core in [0.5,2.0]
    mlp_matvec_wmma<<<grid_val, blk, 0, stream>>>(w2_val, h_val, b2_val,
                                                  y_hat, N_SAMPLES, N_SAMPLES,
                                                  1, -1.0f, 1.0f);
    mlp_matvec_wmma<<<grid_off, blk, 0, stream>>>(w2_off, h_off, b2_off,
                                                  off_core, N_CORE, N_CORE,
                                                  1, 0.5f, 2.0f);
    // x_hat = cumsum(offsets), x_hat[-1] = seqlen-1
    build_xhat<<<1, 256, 0, stream>>>(off_core, x_hat, (float)(N_NEW - 1));
    // y_pred = interp1d(x_hat, y_hat, x_new)
    const int interp_blocks = (N_NEW + 1023) / 1024;
    interp1d_kernel<<<interp_blocks, 256, 0, stream>>>(x_new, x_hat, y_hat,
                                                       y_pred, N_NEW);
}